// ScaledDotProductAttention_47382079209993
// MI455X (gfx1250) — compile-verified
//
#include <hip/hip_runtime.h>

typedef float v2f __attribute__((ext_vector_type(2)));
typedef float v8f __attribute__((ext_vector_type(8)));

#define BATCH 4
#define LSEQ  4096
#define DDIM  64
#define QTILE 16
#define ROWSTRIDE 4100            // 4096 + 4 pad -> conflict-free LDS banks
#define NWAVE 8
#define EPSW  1e-7f
#define NEGINF (-3.402823466e38f)

__launch_bounds__(256)
__global__ void sparse_attn_kernel(const float* __restrict__ Q,
                                   const float* __restrict__ K,
                                   const float* __restrict__ V,
                                   float* __restrict__ Out,
                                   float* __restrict__ Attn)
{
    extern __shared__ float smem[];            // [QTILE][ROWSTRIDE] scores
    const int tid  = threadIdx.x;
    const int wave = tid >> 5;
    const int lane = tid & 31;
    const int b    = blockIdx.y;
    const int q0   = blockIdx.x * QTILE;

    const int mrow = lane & 15;                // M (A) / N (B) row within tile
    const int kofs = (lane >> 4) << 1;         // lanes 16..31 hold K+2,K+3

    // ---------------- phase 1: S = (Q/8) K^T via WMMA f32 16x16x4 ----------
    const float* qrow = Q + ((size_t)(b * LSEQ + q0 + mrow)) * DDIM;
    v2f afrag[16];
#pragma unroll
    for (int t = 0; t < 16; ++t) {
        const float* p = qrow + t * 4 + kofs;
        afrag[t].x = p[0] * 0.125f;            // fold 1/temperature into A
        afrag[t].y = p[1] * 0.125f;
    }

    const float* kbase = K + ((size_t)b * LSEQ) * DDIM;
    for (int kt = wave; kt < LSEQ / 16; kt += NWAVE) {
        const int n0 = kt * 16;
        const float* krow = kbase + ((size_t)(n0 + mrow)) * DDIM;
        v8f c = {0.f, 0.f, 0.f, 0.f, 0.f, 0.f, 0.f, 0.f};
#pragma unroll
        for (int t = 0; t < 16; ++t) {
            v2f bfrag;
            const float* p = krow + t * 4 + kofs;
            bfrag.x = p[0];
            bfrag.y = p[1];
            c = __builtin_amdgcn_wmma_f32_16x16x4_f32(
                    false, afrag[t], false, bfrag,
                    (short)0, c, false, false);
        }
        // C/D layout: lanes 0-15 -> M=vgpr, lanes 16-31 -> M=vgpr+8, N=lane&15
        const int mbase = (lane >> 4) << 3;
#pragma unroll
        for (int vv = 0; vv < 8; ++vv)
            smem[(size_t)(mbase + vv) * ROWSTRIDE + n0 + (lane & 15)] = c[vv];
    }

    __syncthreads();

    // ---------------- phase 2: softmax + top-5 sparsify per row ------------
#pragma unroll
    for (int rr = 0; rr < 2; ++rr) {
        const int m = wave * 2 + rr;
        const float* row = smem + (size_t)m * ROWSTRIDE;

        // row max
        float mx = NEGINF;
        for (int n = lane; n < LSEQ; n += 32) mx = fmaxf(mx, row[n]);
#pragma unroll
        for (int off = 16; off >= 1; off >>= 1)
            mx = fmaxf(mx, __shfl_xor(mx, off, 32));

        // sum(exp) + per-lane local top-5 (values strictly ordered, ties keep
        // the earlier/lower index, matching lax.top_k)
        float se = 0.f;
        float tv0 = NEGINF, tv1 = NEGINF, tv2 = NEGINF, tv3 = NEGINF, tv4 = NEGINF;
        int   ti0 = -1, ti1 = -1, ti2 = -1, ti3 = -1, ti4 = -1;
        for (int n = lane; n < LSEQ; n += 32) {
            float s = row[n];
            se += __expf(s - mx);
            if (s > tv4) {
                if (s > tv0) {
                    tv4=tv3; ti4=ti3; tv3=tv2; ti3=ti2; tv2=tv1; ti2=ti1;
                    tv1=tv0; ti1=ti0; tv0=s; ti0=n;
                } else if (s > tv1) {
                    tv4=tv3; ti4=ti3; tv3=tv2; ti3=ti2; tv2=tv1; ti2=ti1;
                    tv1=s; ti1=n;
                } else if (s > tv2) {
                    tv4=tv3; ti4=ti3; tv3=tv2; ti3=ti2; tv2=s; ti2=n;
                } else if (s > tv3) {
                    tv4=tv3; ti4=ti3; tv3=s; ti3=n;
                } else {
                    tv4=s; ti4=n;
                }
            }
        }
#pragma unroll
        for (int off = 16; off >= 1; off >>= 1) se += __shfl_xor(se, off, 32);

        // cross-lane merge: 5 rounds of wave arg-max; winner pops its head
        float topv[5]; int topi[5];
#pragma unroll
        for (int r = 0; r < 5; ++r) {
            float vbest = tv0; int ibest = ti0;
#pragma unroll
            for (int off = 16; off >= 1; off >>= 1) {
                float ov = __shfl_xor(vbest, off, 32);
                int   oi = __shfl_xor(ibest, off, 32);
                if (ov > vbest || (ov == vbest && (unsigned)oi < (unsigned)ibest)) {
                    vbest = ov; ibest = oi;
                }
            }
            topv[r] = vbest; topi[r] = ibest;
            if (ibest == ti0) {                 // key indices unique per lane
                tv0=tv1; ti0=ti1; tv1=tv2; ti1=ti2; tv2=tv3; ti2=ti3;
                tv3=tv4; ti3=ti4; tv4=NEGINF; ti4=-1;
            }
        }

        const float inv_se = 1.f / se;
        const float delta  = __expf(topv[4] - mx) * inv_se + EPSW;

        float w[5];
        float wsum = 0.f;
#pragma unroll
        for (int r = 0; r < 5; ++r) {
            w[r] = fmaxf(__expf(topv[r] - mx) * inv_se - delta, 0.f);
            wsum += w[r];
        }
        const float invw = 1.f / (wsum + EPSW);

        // out[m,:] = sum of <=4 surviving candidates (exact: every entry with
        // attn > delta is strictly above the 5th largest, hence in topv[0..3])
        float acc0 = 0.f, acc1 = 0.f;
#pragma unroll
        for (int r = 0; r < 5; ++r) {
            float wn = w[r] * invw;
            if (wn > 0.f) {
                const float* vr = V + ((size_t)(b * LSEQ + topi[r])) * DDIM;
                acc0 += wn * vr[lane];
                acc1 += wn * vr[lane + 32];
            }
        }
        float* orow = Out + ((size_t)(b * LSEQ + q0 + m)) * DDIM;
        orow[lane]      = acc0;
        orow[lane + 32] = acc1;

        // full sparse attn row write (float4, coalesced)
        float* arow = Attn + ((size_t)(b * LSEQ + q0 + m)) * (size_t)LSEQ;
        for (int n0 = lane * 4; n0 < LSEQ; n0 += 128) {
            float4 sv = *(const float4*)(row + n0);
            float4 wv;
            wv.x = fmaxf(__expf(sv.x - mx) * inv_se - delta, 0.f) * invw;
            wv.y = fmaxf(__expf(sv.y - mx) * inv_se - delta, 0.f) * invw;
            wv.z = fmaxf(__expf(sv.z - mx) * inv_se - delta, 0.f) * invw;
            wv.w = fmaxf(__expf(sv.w - mx) * inv_se - delta, 0.f) * invw;
            *(float4*)(arow + n0) = wv;
        }
    }
}

extern "C" void kernel_launch(void* const* d_in, const int* in_sizes, int n_in,
                              void* d_out, int out_size, void* d_ws, size_t ws_size,
                              hipStream_t stream) {
    (void)in_sizes; (void)n_in; (void)d_ws; (void)ws_size; (void)out_size;
    const float* q = (const float*)d_in[0];
    const float* k = (const float*)d_in[1];
    const float* v = (const float*)d_in[2];
    float* out  = (float*)d_out;
    float* attn = out + (size_t)BATCH * LSEQ * DDIM;   // tuple: (out, attn)

    dim3 grid(LSEQ / QTILE, BATCH);                    // 256 x 4 workgroups
    dim3 block(256);                                   // 8 wave32
    size_t shmem = (size_t)QTILE * ROWSTRIDE * sizeof(float);  // 262,400 B LDS
    sparse_attn_kernel<<<grid, block, shmem, stream>>>(q, k, v, out, attn);
}